// ClaimValidationLoss_67611375173824
// MI455X (gfx1250) — compile-verified
//
#include <hip/hip_runtime.h>
#include <hip/hip_bf16.h>

// Problem constants (match reference)
#define BATCH   256
#define NVARS   512
#define MCLAIMS 1024
#define EPSV    1e-7f

typedef __attribute__((ext_vector_type(2))) float v2f;
typedef __attribute__((ext_vector_type(8))) float v8f;

// ds_swizzle group-of-32 immediates: offset[14:10]=xor, [9:5]=or, [4:0]=and
#define SWZ_XOR1  0x041F  // xor=1,  and=0x1f
#define SWZ_XOR2  0x081F  // xor=2,  and=0x1f
#define SWZ_XOR4  0x101F  // xor=4,  and=0x1f
#define SWZ_XOR8  0x201F  // xor=8,  and=0x1f
#define SWZ_BC0   0x0000  // and=0, or=0  -> broadcast lane 0
#define SWZ_BC16  0x0200  // and=0, or=16 -> broadcast lane 16

__device__ __forceinline__ float swz(float v, int pattern) {
    switch (pattern) {  // resolves to a literal at each call site after inlining
        case SWZ_XOR1: return __int_as_float(__builtin_amdgcn_ds_swizzle(__float_as_int(v), SWZ_XOR1));
        case SWZ_XOR2: return __int_as_float(__builtin_amdgcn_ds_swizzle(__float_as_int(v), SWZ_XOR2));
        case SWZ_XOR4: return __int_as_float(__builtin_amdgcn_ds_swizzle(__float_as_int(v), SWZ_XOR4));
        case SWZ_XOR8: return __int_as_float(__builtin_amdgcn_ds_swizzle(__float_as_int(v), SWZ_XOR8));
        case SWZ_BC0:  return __int_as_float(__builtin_amdgcn_ds_swizzle(__float_as_int(v), SWZ_BC0));
        default:       return __int_as_float(__builtin_amdgcn_ds_swizzle(__float_as_int(v), SWZ_BC16));
    }
}

// Dual cross-wave reduction using one v_wmma_f32_16x16x4_f32.
//
// B (4x16, 2 VGPRs): b.x = K=0 row (lanes 0-15) / K=2 row (lanes 16-31) -> loss partials
//                    b.y = K=1 row / K=3 row                            -> count partials
// A (16x4, 2 VGPRs): A[m][k] = (m<8) ? (k even) : (k odd)
//                    => lane layout: a.x = (lane%16 < 8), a.y = 1 - a.x
// D[m][n]: m<8  -> B[0][n]+B[2][n] = sum[n]+sum[n+16]
//          m>=8 -> B[1][n]+B[3][n] = cnt[n]+cnt[n+16]
// d[0] holds (M=0,N=lane) for lanes 0-15 (loss colsums) and (M=8,N=lane-16)
// for lanes 16-31 (count colsums). Four xor-swizzles (masks < 16 stay inside
// each half) finish: lane 0 -> total loss sum, lane 16 -> total valid count.
__device__ __forceinline__ float wmma_dual_reduce(float acc_sum, float acc_cnt) {
    const int lane = threadIdx.x & 31;
    v2f a, b;
    a.x = ((lane & 15) < 8) ? 1.0f : 0.0f;
    a.y = 1.0f - a.x;
    b.x = acc_sum;
    b.y = acc_cnt;
    v8f c = {};
    v8f d = __builtin_amdgcn_wmma_f32_16x16x4_f32(
        false, a, false, b, (short)0, c, false, false);
    float v = d[0];
    v += swz(v, SWZ_XOR1);
    v += swz(v, SWZ_XOR2);
    v += swz(v, SWZ_XOR4);
    v += swz(v, SWZ_XOR8);
    return v;  // lane 0: total sum, lane 16: total count
}

// Stage 1: one block per batch row; 256 threads x 4 contiguous claims each.
// Index arrays: one b128 load per array per thread. Adjacency gather: scattered,
// single-use, dataset (256 MB) > L2 (192 MB) -> non-temporal, and predicated on
// rel<4 (the only divergence we keep: it skips ~20% of scattered HBM traffic).
__global__ __launch_bounds__(256) void claim_loss_partial(
    const float* __restrict__ adj,         // [B, N, N]
    const int* __restrict__ var_a,         // [B, M]
    const int* __restrict__ var_b,         // [B, M]
    const int* __restrict__ rel,           // [B, M]
    const int* __restrict__ is_true,       // [B, M]
    const unsigned char* __restrict__ msk, // [B, M] bool, True = padding
    float* __restrict__ part_sum,          // [B]
    float* __restrict__ part_cnt)          // [B]
{
    const int b = blockIdx.x;
    const int t = threadIdx.x;
    const size_t adj_base = (size_t)b * NVARS * NVARS;
    const int row = b * MCLAIMS;

    // Vectorized coalesced index loads (rows are 4 KB aligned).
    const int4 A4 = ((const int4*)(var_a   + row))[t];
    const int4 B4 = ((const int4*)(var_b   + row))[t];
    const int4 R4 = ((const int4*)(rel     + row))[t];
    const int4 T4 = ((const int4*)(is_true + row))[t];
    const uchar4 M4 = ((const uchar4*)(msk + row))[t];

    const int va[4] = {A4.x, A4.y, A4.z, A4.w};
    const int vb[4] = {B4.x, B4.y, B4.z, B4.w};
    const int rr[4] = {R4.x, R4.y, R4.z, R4.w};
    const int tt[4] = {T4.x, T4.y, T4.z, T4.w};
    const unsigned char mm[4] = {M4.x, M4.y, M4.z, M4.w};

    // Issue all gathers up front (rel>=4 never touches memory).
    float pv[4];
#pragma unroll
    for (int i = 0; i < 4; ++i) {
        pv[i] = 0.5f;
        if (rr[i] < 4)
            pv[i] = __builtin_nontemporal_load(
                adj + adj_base + (size_t)va[i] * NVARS + vb[i]);
    }

    float lsum = 0.0f, lcnt = 0.0f;
#pragma unroll
    for (int i = 0; i < 4; ++i) {
        // rel∈[0,4]: flip ⇔ rel∈{1,3} ⇔ rel is odd -> single bit test.
        const bool flip = (rr[i] & 1) != 0;
        float prob = flip ? (1.0f - pv[i]) : pv[i];
        prob = fminf(fmaxf(prob, EPSV), 1.0f - EPSV);
        // Compute both log branches (cheap TRANS ops), select branchlessly.
        const float lp  = -logf(prob);       // t = 1 term
        const float lq  = -log1pf(-prob);    // t = 0 term (reference uses log1p)
        const float per = tt[i] ? lp : lq;
        const float valid = mm[i] ? 0.0f : 1.0f;
        lsum = fmaf(per, valid, lsum);
        lcnt += valid;
    }

    __shared__ float s_sum[256];
    __shared__ float s_cnt[256];
    s_sum[t] = lsum;
    s_cnt[t] = lcnt;
    __syncthreads();

    if (t < 32) {  // wave 0 fully active -> EXEC all-ones for the WMMA
        float as = 0.0f, ac = 0.0f;
#pragma unroll
        for (int i = 0; i < 8; ++i) {
            as += s_sum[t + 32 * i];
            ac += s_cnt[t + 32 * i];
        }
        const float v = wmma_dual_reduce(as, ac);
        if (t == 0)  part_sum[b] = v;
        if (t == 16) part_cnt[b] = v;
    }
}

// Stage 2: a single wave folds the 256 per-block partials in fixed order
// (bitwise deterministic across graph replays), then writes the masked mean.
__global__ __launch_bounds__(32) void claim_loss_final(
    const float* __restrict__ part_sum,
    const float* __restrict__ part_cnt,
    float* __restrict__ out)
{
    const int t = threadIdx.x;  // 32 threads, EXEC all-ones
    float as = 0.0f, ac = 0.0f;
#pragma unroll
    for (int i = 0; i < BATCH / 32; ++i) {
        as += part_sum[t + 32 * i];
        ac += part_cnt[t + 32 * i];
    }
    const float v = wmma_dual_reduce(as, ac);
    const float total_sum = swz(v, SWZ_BC0);
    const float total_cnt = swz(v, SWZ_BC16);
    if (t == 0) {
        out[0] = (total_cnt > 0.0f) ? (total_sum / fmaxf(total_cnt, 1.0f)) : 0.0f;
    }
}

extern "C" void kernel_launch(void* const* d_in, const int* in_sizes, int n_in,
                              void* d_out, int out_size, void* d_ws, size_t ws_size,
                              hipStream_t stream) {
    const float*         adj = (const float*)d_in[0];
    const int*           va  = (const int*)d_in[1];
    const int*           vb  = (const int*)d_in[2];
    const int*           rl  = (const int*)d_in[3];
    const int*           it  = (const int*)d_in[4];
    const unsigned char* mk  = (const unsigned char*)d_in[5];
    float*               out = (float*)d_out;

    float* part_sum = (float*)d_ws;      // BATCH floats
    float* part_cnt = part_sum + BATCH;  // BATCH floats (2 KB total)

    claim_loss_partial<<<BATCH, 256, 0, stream>>>(adj, va, vb, rl, it, mk,
                                                  part_sum, part_cnt);
    claim_loss_final<<<1, 32, 0, stream>>>(part_sum, part_cnt, out);
}